// PostProcessor_25975962206916
// MI455X (gfx1250) — compile-verified
//
#include <hip/hip_runtime.h>
#include <math.h>

#define NPROP 1024
#define NCLS  81
#define KCLS  80
#define NEG_INF (-INFINITY)

#define ROWS_PER_BLK 128   // k_softmax_decode rows per block (128 threads)

__device__ __forceinline__ float clampf(float v, float lo, float hi) {
    return fminf(fmaxf(v, lo), hi);
}

// ---------------------------------------------------------------------------
// Kernel 1: per-proposal softmax + box decode (maskrcnn BoxCoder, TO_REMOVE=1).
// Logits are staged block-wide into LDS with coalesced async global->LDS B128
// copies (ASYNCcnt), then each thread reduces its row out of LDS (stride-81
// dword access across lanes -> bank-conflict free). Regression is read with
// aligned float4 loads. Writes scores [K][N] (-inf below thresh), boxes [K][N][4].
// ---------------------------------------------------------------------------
__global__ void k_softmax_decode(const float* __restrict__ logits,
                                 const float* __restrict__ reg,
                                 const float* __restrict__ refb,
                                 const int* __restrict__ hptr,
                                 const int* __restrict__ wptr,
                                 float* __restrict__ sc_un,    // [K][N]
                                 float* __restrict__ bx_dec)   // [K][N][4]
{
    __shared__ float lds_logits[ROWS_PER_BLK * NCLS];  // 40.5 KB

    const int t  = threadIdx.x;
    const int r0 = blockIdx.x * ROWS_PER_BLK;

    // ---- coalesced async staging of this block's logits (ROWS*81 floats) ----
    {
        const float* src = logits + (size_t)r0 * NCLS;
        const int n16 = (ROWS_PER_BLK * NCLS * 4) / 16;  // # of 16B chunks
        for (int f = t; f < n16; f += ROWS_PER_BLK) {
            unsigned voff = (unsigned)f * 16u;
            unsigned lds  = (unsigned)(uintptr_t)(&lds_logits[f * 4]);
            asm volatile("global_load_async_to_lds_b128 %0, %1, %2"
                         :: "v"(lds), "v"(voff), "s"(src) : "memory");
        }
        asm volatile("s_wait_asynccnt 0" ::: "memory");
        __syncthreads();
    }

    const int r = r0 + t;
    const float* l = &lds_logits[t * NCLS];

    float mx = l[0];
    for (int c = 1; c < NCLS; ++c) mx = fmaxf(mx, l[c]);
    float sum = 0.0f;
    for (int c = 0; c < NCLS; ++c) sum += expf(l[c] - mx);
    float inv = 1.0f / sum;

    float rx1 = refb[r * 4 + 0], ry1 = refb[r * 4 + 1];
    float rx2 = refb[r * 4 + 2], ry2 = refb[r * 4 + 3];
    float w  = rx2 - rx1 + 1.0f, h  = ry2 - ry1 + 1.0f;
    float cx = rx1 + 0.5f * w,   cy = ry1 + 0.5f * h;
    float Wm1 = (float)(*wptr) - 1.0f;
    float Hm1 = (float)(*hptr) - 1.0f;
    const float CLIP = 4.135166556742356f;  // log(1000/16)

    for (int c = 1; c < NCLS; ++c) {
        float prob = expf(l[c] - mx) * inv;
        float s = (prob > 0.05f) ? prob : NEG_INF;

        // 16B-aligned vector load of [dx,dy,dw,dh]
        const float4 rr = *(const float4*)(reg + (size_t)r * (NCLS * 4) + c * 4);
        float dx = rr.x / 10.0f;
        float dy = rr.y / 10.0f;
        float dw = fminf(rr.z / 5.0f, CLIP);
        float dh = fminf(rr.w / 5.0f, CLIP);
        float pcx = dx * w + cx, pcy = dy * h + cy;
        float pw = expf(dw) * w, ph = expf(dh) * h;
        float x1 = clampf(pcx - 0.5f * pw,        0.0f, Wm1);
        float y1 = clampf(pcy - 0.5f * ph,        0.0f, Hm1);
        float x2 = clampf(pcx + 0.5f * pw - 1.0f, 0.0f, Wm1);
        float y2 = clampf(pcy + 0.5f * ph - 1.0f, 0.0f, Hm1);

        int k = c - 1;
        sc_un[k * NPROP + r] = s;
        float4* bp = (float4*)(bx_dec + ((size_t)k * NPROP + r) * 4);
        *bp = make_float4(x1, y1, x2, y2);
    }
}

// ---------------------------------------------------------------------------
// Kernel 2: per-class descending bitonic sort (1024 elems, 1024 threads, LDS).
// Emits sorted scores + permutation indices (boxes gathered lazily later).
// ---------------------------------------------------------------------------
__global__ void k_sort(const float* __restrict__ sc_un,
                       float* __restrict__ sc_sorted,
                       int* __restrict__ idx_sorted)
{
    __shared__ float key[NPROP];
    __shared__ int   val[NPROP];
    const int k = blockIdx.x;
    const int t = threadIdx.x;

    key[t] = sc_un[k * NPROP + t];
    val[t] = t;

    for (int size = 2; size <= NPROP; size <<= 1) {
        for (int stride = size >> 1; stride > 0; stride >>= 1) {
            __syncthreads();
            int j = t ^ stride;
            if (j > t) {
                float a = key[t], b = key[j];
                bool descBlk = ((t & size) == 0);
                bool sw = descBlk ? (a < b) : (a > b);
                if (sw) {
                    key[t] = b; key[j] = a;
                    int va = val[t]; val[t] = val[j]; val[j] = va;
                }
            }
        }
    }
    __syncthreads();
    sc_sorted[k * NPROP + t]  = key[t];
    idx_sorted[k * NPROP + t] = val[t];
}

// ---------------------------------------------------------------------------
// Kernel 3: per-class greedy NMS over the valid (score > -inf) sorted prefix.
// Boxes gathered into LDS with CDNA5 async global->LDS B128 loads (ASYNCcnt).
// Suppressed entries get -inf written back into sc_sorted.
// ---------------------------------------------------------------------------
__global__ void k_nms(float* __restrict__ sc_sorted,
                      const int* __restrict__ idx_sorted,
                      const float* __restrict__ bx_dec)
{
    __shared__ float bxs[NPROP * 4];   // 16 KB
    __shared__ float area[NPROP];      //  4 KB
    __shared__ int   keep[NPROP];      //  4 KB
    __shared__ int   Vsh;

    const int k = blockIdx.x;
    const int t = threadIdx.x;
    const int T = blockDim.x;

    if (t == 0) Vsh = NPROP;
    __syncthreads();
    for (int i = t; i < NPROP; i += T)
        if (sc_sorted[k * NPROP + i] == NEG_INF) atomicMin(&Vsh, i);
    __syncthreads();
    const int V = Vsh;   // sorted desc => valid prefix [0, V)

    if (V > 0) {
        const float* base = bx_dec + (size_t)k * NPROP * 4;
        // Async indirect gather: each lane issues a 16B global->LDS copy of
        // its permuted box row (GVS addressing: SGPR base + lane byte offset).
        for (int i = t; i < V; i += T) {
            keep[i] = 1;
            unsigned row  = (unsigned)idx_sorted[k * NPROP + i];
            unsigned voff = row * 16u;
            unsigned lds  = (unsigned)(uintptr_t)(&bxs[i * 4]);  // low 32b = LDS addr
            asm volatile("global_load_async_to_lds_b128 %0, %1, %2"
                         :: "v"(lds), "v"(voff), "s"(base) : "memory");
        }
        asm volatile("s_wait_asynccnt 0" ::: "memory");
        __syncthreads();

        for (int i = t; i < V; i += T) {
            float x1 = bxs[4*i+0], y1 = bxs[4*i+1];
            float x2 = bxs[4*i+2], y2 = bxs[4*i+3];
            area[i] = (x2 - x1 + 1.0f) * (y2 - y1 + 1.0f);
        }
        __syncthreads();

        for (int i = 0; i < V; ++i) {
            if (keep[i]) {                       // uniform: stable after barrier
                float xi1 = bxs[4*i+0], yi1 = bxs[4*i+1];
                float xi2 = bxs[4*i+2], yi2 = bxs[4*i+3];
                float ai  = area[i];
                for (int j = i + 1 + t; j < V; j += T) {
                    if (keep[j]) {
                        float iw = fminf(xi2, bxs[4*j+2]) - fmaxf(xi1, bxs[4*j+0]) + 1.0f;
                        float ih = fminf(yi2, bxs[4*j+3]) - fmaxf(yi1, bxs[4*j+1]) + 1.0f;
                        iw = fmaxf(iw, 0.0f); ih = fmaxf(ih, 0.0f);
                        float inter = iw * ih;
                        float iou = inter / (ai + area[j] - inter);
                        if (iou > 0.5f) keep[j] = 0;
                    }
                }
            }
            __syncthreads();
        }

        for (int i = t; i < V; i += T)
            if (!keep[i]) sc_sorted[k * NPROP + i] = NEG_INF;
    }
}

// ---------------------------------------------------------------------------
// Kernel 4: global top-100 over the 80*1024 flat scores, index tie-break to
// match lax.top_k ordering. wave32 shuffle reduction (5 shfl steps) + one
// cross-wave LDS stage per pass instead of a 10-barrier tree.
// ---------------------------------------------------------------------------
__global__ void k_topk(float* __restrict__ sc_sorted,
                       const int* __restrict__ idx_sorted,
                       const float* __restrict__ bx_dec,
                       float* __restrict__ out)
{
    __shared__ float wv[32];
    __shared__ int   wi[32];
    const int t    = threadIdx.x;
    const int wid  = t >> 5;
    const int lane = t & 31;
    const int M    = KCLS * NPROP;

    for (int pass = 0; pass < 100; ++pass) {
        float best = NEG_INF; int bi = M;
        for (int idx = t; idx < M; idx += 1024) {
            float v = sc_sorted[idx];
            if (v > best) { best = v; bi = idx; }   // ascending scan => lowest idx on ties
        }
        // intra-wave (wave32) reduction
        for (int off = 16; off > 0; off >>= 1) {
            float vo = __shfl_xor(best, off, 32);
            int   io = __shfl_xor(bi,   off, 32);
            if (vo > best || (vo == best && io < bi)) { best = vo; bi = io; }
        }
        if (lane == 0) { wv[wid] = best; wi[wid] = bi; }
        __syncthreads();
        if (t < 32) {
            best = wv[t]; bi = wi[t];
            for (int off = 16; off > 0; off >>= 1) {
                float vo = __shfl_xor(best, off, 32);
                int   io = __shfl_xor(bi,   off, 32);
                if (vo > best || (vo == best && io < bi)) { best = vo; bi = io; }
            }
            if (t == 0) {
                if (best > NEG_INF && bi < M) {
                    int kcl = bi >> 10;          // class index (0..79)
                    int row = idx_sorted[bi];    // original proposal row
                    const float* bp = bx_dec + ((size_t)kcl * NPROP + row) * 4;
                    out[pass]               = best;
                    out[100 + pass * 4 + 0] = bp[0];
                    out[100 + pass * 4 + 1] = bp[1];
                    out[100 + pass * 4 + 2] = bp[2];
                    out[100 + pass * 4 + 3] = bp[3];
                    out[500 + pass]         = (float)(kcl + 1);
                    sc_sorted[bi] = NEG_INF;     // remove from candidate set
                } else {
                    out[pass] = 0.0f;
                    out[100 + pass * 4 + 0] = 0.0f;
                    out[100 + pass * 4 + 1] = 0.0f;
                    out[100 + pass * 4 + 2] = 0.0f;
                    out[100 + pass * 4 + 3] = 0.0f;
                    out[500 + pass] = 0.0f;
                }
            }
        }
        __syncthreads();   // make thread0's global write visible to next pass
    }
}

// ---------------------------------------------------------------------------
// Launcher. Workspace layout (bytes):
//   [0, 320K)            scores unsorted [K][N] f32
//   [320K, 320K+1.25M)   decoded boxes   [K][N][4] f32
//   next 320K            scores sorted   [K][N] f32   (mutated by NMS + topk)
//   next 320K            sort permutation[K][N] i32
// Total ~2.24 MB. Every region is fully rewritten each call -> deterministic.
// ---------------------------------------------------------------------------
extern "C" void kernel_launch(void* const* d_in, const int* in_sizes, int n_in,
                              void* d_out, int out_size, void* d_ws, size_t ws_size,
                              hipStream_t stream)
{
    const float* logits = (const float*)d_in[0];
    const float* reg    = (const float*)d_in[1];
    const float* refb   = (const float*)d_in[2];
    const int*   hptr   = (const int*)d_in[3];
    const int*   wptr   = (const int*)d_in[4];
    float* out = (float*)d_out;

    char* ws = (char*)d_ws;
    const size_t SC_BYTES = (size_t)KCLS * NPROP * sizeof(float);      // 320 KB
    const size_t BX_BYTES = (size_t)KCLS * NPROP * 4 * sizeof(float);  // 1.25 MB
    float* sc_un      = (float*)(ws);
    float* bx_dec     = (float*)(ws + SC_BYTES);
    float* sc_sorted  = (float*)(ws + SC_BYTES + BX_BYTES);
    int*   idx_sorted = (int*)  (ws + SC_BYTES + BX_BYTES + SC_BYTES);

    k_softmax_decode<<<NPROP / ROWS_PER_BLK, ROWS_PER_BLK, 0, stream>>>(
        logits, reg, refb, hptr, wptr, sc_un, bx_dec);
    k_sort<<<KCLS, NPROP, 0, stream>>>(sc_un, sc_sorted, idx_sorted);
    k_nms<<<KCLS, 256, 0, stream>>>(sc_sorted, idx_sorted, bx_dec);
    k_topk<<<1, 1024, 0, stream>>>(sc_sorted, idx_sorted, bx_dec, out);
}